// UnifiedGATModel_29789893165161
// MI455X (gfx1250) — compile-verified
//
#include <hip/hip_runtime.h>
#include <math.h>
#include <stdint.h>

typedef __attribute__((ext_vector_type(2))) float v2f;
typedef __attribute__((ext_vector_type(8))) float v8f;

#define NFULL 50000
#define NGRP  512
#define DH    128
#define NCLS  16
#define NEG_SLOPE 0.2f

// ---------- helpers ----------

__device__ __forceinline__ void atomicMaxF(float* addr, float val) {
    int old = __float_as_int(*addr);
    while (val > __int_as_float(old)) {
        int assumed = old;
        old = atomicCAS((int*)addr, assumed, __float_as_int(val));
        if (old == assumed) break;
    }
}

__device__ __forceinline__ void edge_sd(const int* __restrict__ src,
                                        const int* __restrict__ dst,
                                        int E, int e, int& s, int& d) {
    if (e < E) { s = src[e]; d = dst[e]; }
    else       { s = e - E; d = e - E; }   // self loop
}

// ---------- WMMA GEMM: C[M x ncolsB] = A[M x 128] * B[128 x ncolsB] (+ bias) ----------
// f32 path: V_WMMA_F32_16X16X4_F32, K-loop of 32 chained wmma ops per 16x16 tile.
// Dense B panel (ldB == ncolsB) staged memory->LDS with GLOBAL_LOAD_ASYNC_TO_LDS_B128
// (ASYNCcnt path, no VGPR round-trip), then s_wait_asynccnt 0 + barrier.
// One wave per output tile; whole-wave early exit keeps EXEC all-ones for every WMMA.

__global__ void gemm_wmma_kernel(const float* __restrict__ A,
                                 const float* __restrict__ B,
                                 const float* __restrict__ bias,
                                 float* __restrict__ C,
                                 int rowTiles, int colTiles,
                                 int ldA, int ldC, int ncolsB) {
    __shared__ float sB[128 * 128];   // 64 KB max: full B panel (K=128, up to 128 cols)
    int tid = threadIdx.x;

    // async stage: 16-byte chunks, per-lane LDS offset + per-lane global address
    int nChunks = (128 * ncolsB) >> 2;         // #(4-float chunks)
    for (int i = tid; i < nChunks; i += blockDim.x) {
        uint32_t ldsOff = (uint32_t)(uintptr_t)(&sB[i << 2]);   // low 32 bits = LDS offset
        const float* gsrc = B + ((long)i << 2);
        asm volatile("global_load_async_to_lds_b128 %0, %1, off"
                     :: "v"(ldsOff), "v"(gsrc) : "memory");
    }
    asm volatile("s_wait_asynccnt 0" ::: "memory");
    __syncthreads();

    int wave = blockIdx.x * (blockDim.x >> 5) + (tid >> 5);
    int nTiles = rowTiles * colTiles;
    if (wave >= nTiles) return;               // whole wave exits together

    int tm   = wave / colTiles;
    int tn   = wave % colTiles;
    int lane = tid & 31;
    int l15  = lane & 15;
    int khalf = (lane >> 4) << 1;             // 0 for lanes 0-15, 2 for lanes 16-31
    int row  = tm * 16 + l15;                 // A-frag: M = lane&15
    int col  = tn * 16 + l15;                 // B/C-frag: N = lane&15

    v8f c = {0.f, 0.f, 0.f, 0.f, 0.f, 0.f, 0.f, 0.f};
    const float* arow = A + (long)row * ldA;

    #pragma unroll 4
    for (int k0 = 0; k0 < 128; k0 += 4) {
        int k = k0 + khalf;
        v2f a, b;
        a.x = arow[k];
        a.y = arow[k + 1];
        b.x = sB[k * ncolsB + col];
        b.y = sB[(k + 1) * ncolsB + col];
        c = __builtin_amdgcn_wmma_f32_16x16x4_f32(
                /*neg_a=*/false, a, /*neg_b=*/false, b,
                /*c_mod=*/(short)0, c, /*reuse_a=*/false, /*reuse_b=*/false);
    }

    int rbase = tm * 16 + ((lane >> 4) << 3);     // lanes 16-31 hold M = 8..15
    float bc = bias ? bias[col] : 0.f;
    #pragma unroll
    for (int v = 0; v < 8; ++v) {
        C[(long)(rbase + v) * ldC + col] = c[v] + bc;
    }
}

// ---------- misc kernels ----------

__global__ void zero_kernel(float* __restrict__ p, long n) {
    long i = (long)blockIdx.x * blockDim.x + threadIdx.x;
    if (i < n) p[i] = 0.f;
}

// per-node attention scores + softmax-state init
__global__ void attn_scores_kernel(const float* __restrict__ h,
                                   const float* __restrict__ a_src,
                                   const float* __restrict__ a_dst,
                                   float* __restrict__ as, float* __restrict__ ad,
                                   float* __restrict__ m, float* __restrict__ dn, int n) {
    __shared__ float ss[128], sd[128];
    if (threadIdx.x < 128) {
        ss[threadIdx.x] = a_src[threadIdx.x];
        sd[threadIdx.x] = a_dst[threadIdx.x];
    }
    __syncthreads();
    int i = blockIdx.x * blockDim.x + threadIdx.x;
    if (i >= n) return;
    const float* hr = h + (long)i * DH;
    float s = 0.f, d = 0.f;
    #pragma unroll 8
    for (int k = 0; k < DH; ++k) {
        float hv = hr[k];
        s += hv * ss[k];
        d += hv * sd[k];
    }
    as[i] = s; ad[i] = d;
    m[i] = -INFINITY; dn[i] = 0.f;
}

__global__ void edge_max_kernel(const int* __restrict__ src, const int* __restrict__ dst,
                                int E, int n,
                                const float* __restrict__ as, const float* __restrict__ ad,
                                float* __restrict__ m) {
    int e = blockIdx.x * blockDim.x + threadIdx.x;
    if (e >= E + n) return;
    int s, d; edge_sd(src, dst, E, e, s, d);
    float ev = as[s] + ad[d];
    ev = ev > 0.f ? ev : NEG_SLOPE * ev;
    atomicMaxF(&m[d], ev);
}

__global__ void edge_expsum_kernel(const int* __restrict__ src, const int* __restrict__ dst,
                                   int E, int n,
                                   const float* __restrict__ as, const float* __restrict__ ad,
                                   const float* __restrict__ m, float* __restrict__ dn) {
    int e = blockIdx.x * blockDim.x + threadIdx.x;
    if (e >= E + n) return;
    int s, d; edge_sd(src, dst, E, e, s, d);
    float ev = as[s] + ad[d];
    ev = ev > 0.f ? ev : NEG_SLOPE * ev;
    atomicAdd(&dn[d], __expf(ev - m[d]));
}

// wave per edge: agg[dst] += alpha * h[src], 32 lanes x 4 features
__global__ void edge_scatter_kernel(const int* __restrict__ src, const int* __restrict__ dst,
                                    int E, int n,
                                    const float* __restrict__ as, const float* __restrict__ ad,
                                    const float* __restrict__ m, const float* __restrict__ dn,
                                    const float* __restrict__ h, float* __restrict__ agg) {
    int wave = blockIdx.x * (blockDim.x >> 5) + (threadIdx.x >> 5);
    if (wave >= E + n) return;
    int lane = threadIdx.x & 31;
    int s, d; edge_sd(src, dst, E, wave, s, d);
    float ev = as[s] + ad[d];
    ev = ev > 0.f ? ev : NEG_SLOPE * ev;
    float coef = __expf(ev - m[d]) / dn[d];
    const float* hs = h + (long)s * DH;
    float* ag = agg + (long)d * DH;
    int base = lane * 4;
    #pragma unroll
    for (int t = 0; t < 4; ++t)
        atomicAdd(&ag[base + t], coef * hs[base + t]);
}

__global__ void bias_elu_kernel(float* __restrict__ x, const float* __restrict__ bias, long total) {
    long i = (long)blockIdx.x * blockDim.x + threadIdx.x;
    if (i >= total) return;
    float v = x[i] + bias[i & (DH - 1)];
    x[i] = v > 0.f ? v : __expf(v) - 1.f;
}

// wave per group-edge: adj_sum[gsrc] += group_x[gdst]; cnt[gsrc] += 1
__global__ void adj_accum_kernel(const int* __restrict__ gsrc, const int* __restrict__ gdst,
                                 int E, const float* __restrict__ gx,
                                 float* __restrict__ adj_sum, float* __restrict__ cnt) {
    int wave = blockIdx.x * (blockDim.x >> 5) + (threadIdx.x >> 5);
    if (wave >= E) return;
    int lane = threadIdx.x & 31;
    int s = gsrc[wave], d = gdst[wave];
    const float* g = gx + (long)d * DH;
    float* o = adj_sum + (long)s * DH;
    int base = lane * 4;
    #pragma unroll
    for (int t = 0; t < 4; ++t)
        atomicAdd(&o[base + t], g[base + t]);
    if (lane == 0) atomicAdd(&cnt[s], 1.f);
}

__global__ void adj_mean_kernel(float* __restrict__ adj, const float* __restrict__ cnt, int G) {
    int i = blockIdx.x * blockDim.x + threadIdx.x;
    if (i >= G * DH) return;
    adj[i] /= fmaxf(cnt[i >> 7], 1.f);
}

// per node: updated = f + dot(f,gf)*gf + dot(f,am)*am
__global__ void node_update_kernel(const float* __restrict__ fullx, const float* __restrict__ gx,
                                   const float* __restrict__ adj, const int* __restrict__ node_group,
                                   float* __restrict__ updated, int n) {
    int i = blockIdx.x * blockDim.x + threadIdx.x;
    if (i >= n) return;
    int g = node_group[i];
    const float* f  = fullx + (long)i * DH;
    const float* gf = gx    + (long)g * DH;
    const float* am = adj   + (long)g * DH;
    float d1 = 0.f, d2 = 0.f;
    #pragma unroll 8
    for (int k = 0; k < DH; ++k) { float fv = f[k]; d1 += fv * gf[k]; d2 += fv * am[k]; }
    float* u = updated + (long)i * DH;
    #pragma unroll 8
    for (int k = 0; k < DH; ++k) u[k] = f[k] + d1 * gf[k] + d2 * am[k];
}

// ---------- launcher ----------

extern "C" void kernel_launch(void* const* d_in, const int* in_sizes, int n_in,
                              void* d_out, int out_size, void* d_ws, size_t ws_size,
                              hipStream_t stream) {
    const float* x_full      = (const float*)d_in[0];
    const float* x_group     = (const float*)d_in[1];
    const float* W_full      = (const float*)d_in[2];
    const float* a_src_full  = (const float*)d_in[3];
    const float* a_dst_full  = (const float*)d_in[4];
    const float* bias_full   = (const float*)d_in[5];
    const float* W_group     = (const float*)d_in[6];
    const float* a_src_group = (const float*)d_in[7];
    const float* a_dst_group = (const float*)d_in[8];
    const float* bias_group  = (const float*)d_in[9];
    const float* fc_W        = (const float*)d_in[10];
    const float* fc_b        = (const float*)d_in[11];
    const int*   ei_full     = (const int*)d_in[12];
    const int*   ei_grp      = (const int*)d_in[13];
    const int*   node_group  = (const int*)d_in[14];

    int E_full = in_sizes[12] / 2;
    int E_grp  = in_sizes[13] / 2;
    const int* src_full = ei_full;
    const int* dst_full = ei_full + E_full;
    const int* src_grp  = ei_grp;
    const int* dst_grp  = ei_grp + E_grp;

    // workspace layout (floats); zero-init region first so one zero pass covers it
    float* ws       = (float*)d_ws;
    float* agg_full = ws;                                   // N*128 (attn accum -> full_x)
    float* agg_grp  = agg_full + (long)NFULL * DH;          // G*128 (attn accum -> group_x)
    float* adj_sum  = agg_grp  + (long)NGRP * DH;           // G*128 (-> adj_mean in place)
    float* cnt      = adj_sum  + (long)NGRP * DH;           // G
    float* h_full   = cnt + NGRP;                           // N*128
    float* h_grp    = h_full + (long)NFULL * DH;            // G*128
    float* as_full  = h_grp  + (long)NGRP * DH;             // N
    float* ad_full  = as_full + NFULL;
    float* m_full   = ad_full + NFULL;
    float* dn_full  = m_full  + NFULL;
    float* as_grp   = dn_full + NFULL;                      // G
    float* ad_grp   = as_grp + NGRP;
    float* m_grp    = ad_grp + NGRP;
    float* dn_grp   = m_grp  + NGRP;

    float* out     = (float*)d_out;               // [N,16]
    float* updated = out + (long)NFULL * NCLS;    // [N,128]

    long zcount = (long)NFULL * DH + 2L * NGRP * DH + NGRP;
    zero_kernel<<<(int)((zcount + 255) / 256), 256, 0, stream>>>(ws, zcount);

    // h = x @ W  (WMMA f32 16x16x4; tiles divide exactly: 3125x8 and 32x8)
    gemm_wmma_kernel<<<(NFULL / 16) * 8 / 4, 128, 0, stream>>>(
        x_full, W_full, nullptr, h_full, NFULL / 16, 8, DH, DH, DH);
    gemm_wmma_kernel<<<(NGRP / 16) * 8 / 4, 128, 0, stream>>>(
        x_group, W_group, nullptr, h_grp, NGRP / 16, 8, DH, DH, DH);

    attn_scores_kernel<<<(NFULL + 127) / 128, 128, 0, stream>>>(
        h_full, a_src_full, a_dst_full, as_full, ad_full, m_full, dn_full, NFULL);
    attn_scores_kernel<<<(NGRP + 127) / 128, 128, 0, stream>>>(
        h_grp, a_src_group, a_dst_group, as_grp, ad_grp, m_grp, dn_grp, NGRP);

    int ET_full = E_full + NFULL, ET_grp = E_grp + NGRP;
    edge_max_kernel<<<(ET_full + 255) / 256, 256, 0, stream>>>(
        src_full, dst_full, E_full, NFULL, as_full, ad_full, m_full);
    edge_max_kernel<<<(ET_grp + 255) / 256, 256, 0, stream>>>(
        src_grp, dst_grp, E_grp, NGRP, as_grp, ad_grp, m_grp);

    edge_expsum_kernel<<<(ET_full + 255) / 256, 256, 0, stream>>>(
        src_full, dst_full, E_full, NFULL, as_full, ad_full, m_full, dn_full);
    edge_expsum_kernel<<<(ET_grp + 255) / 256, 256, 0, stream>>>(
        src_grp, dst_grp, E_grp, NGRP, as_grp, ad_grp, m_grp, dn_grp);

    edge_scatter_kernel<<<(ET_full + 7) / 8, 256, 0, stream>>>(
        src_full, dst_full, E_full, NFULL, as_full, ad_full, m_full, dn_full, h_full, agg_full);
    edge_scatter_kernel<<<(ET_grp + 7) / 8, 256, 0, stream>>>(
        src_grp, dst_grp, E_grp, NGRP, as_grp, ad_grp, m_grp, dn_grp, h_grp, agg_grp);

    bias_elu_kernel<<<(int)(((long)NFULL * DH + 255) / 256), 256, 0, stream>>>(
        agg_full, bias_full, (long)NFULL * DH);
    bias_elu_kernel<<<(int)(((long)NGRP * DH + 255) / 256), 256, 0, stream>>>(
        agg_grp, bias_group, (long)NGRP * DH);

    adj_accum_kernel<<<(E_grp + 7) / 8, 256, 0, stream>>>(
        src_grp, dst_grp, E_grp, agg_grp, adj_sum, cnt);
    adj_mean_kernel<<<(NGRP * DH + 255) / 256, 256, 0, stream>>>(adj_sum, cnt, NGRP);

    node_update_kernel<<<(NFULL + 127) / 128, 128, 0, stream>>>(
        agg_full, agg_grp, adj_sum, node_group, updated, NFULL);

    // out = updated @ fc_W + fc_b  (WMMA again; 3125 tiles, last block partially idle)
    gemm_wmma_kernel<<<(NFULL / 16 + 3) / 4, 128, 0, stream>>>(
        updated, fc_W, fc_b, out, NFULL / 16, 1, DH, NCLS, NCLS);
}